// GENNet_19713899889122
// MI455X (gfx1250) — compile-verified
//
#include <hip/hip_runtime.h>
#include <hip/hip_bf16.h>

#define NNODES 50000
#define NEDGES 400000
#define NGRAPH 64

typedef float v2f __attribute__((ext_vector_type(2)));
typedef float v8f __attribute__((ext_vector_type(8)));

// ---------------------------------------------------------------------------
// WMMA f32 GEMM:  C[M,N] = A[M,K] @ W[K,N] + bias
// Block = 256 threads = 8 waves; each wave computes a 16-row x 64-col strip.
// grid.x covers rows (128/block), grid.y covers columns (64/block).
// A-frag (16x4 f32): lane<16 -> row=lane, k,k+1 ; lane>=16 -> row=lane-16, k+2,k+3
// B-frag (4x16 f32): lane<16 -> col=lane, rows k,k+1 ; lane>=16 -> col=lane-16, rows k+2,k+3
// C/D (16x16 f32):   lane<16 -> rows 0..7 ; lane>=16 -> rows 8..15 ; col = lane&15
// ---------------------------------------------------------------------------
template <int K, int N>
__global__ __launch_bounds__(256) void gemm_bias(
    const float* __restrict__ A, const float* __restrict__ W,
    const float* __restrict__ bias, float* __restrict__ C, int M) {
  constexpr int NT = 4;  // 4 x 16 = 64 cols per wave
  const int lane = threadIdx.x & 31;
  const int wave = threadIdx.x >> 5;
  const int halfsel = lane >> 4;
  const int l15 = lane & 15;
  const long row0 = (long)blockIdx.x * 128 + (long)wave * 16;
  const int nbase = blockIdx.y * 64;

  long arow = row0 + l15;
  if (arow >= M) arow = M - 1;  // clamp: garbage rows never stored

  v8f acc[NT];
#pragma unroll
  for (int t = 0; t < NT; ++t)
#pragma unroll
    for (int r = 0; r < 8; ++r) acc[t][r] = 0.0f;

  const float* ap_base = A + arow * K + halfsel * 2;
  for (int k = 0; k < K; k += 4) {
    v2f a;
    a.x = ap_base[k + 0];
    a.y = ap_base[k + 1];
#pragma unroll
    for (int t = 0; t < NT; ++t) {
      const float* wp = W + (long)(k + halfsel * 2) * N + nbase + t * 16 + l15;
      v2f b;
      b.x = wp[0];
      b.y = wp[N];
      acc[t] = __builtin_amdgcn_wmma_f32_16x16x4_f32(
          false, a, false, b, (short)0, acc[t], false, false);
    }
  }

  const int rowadd = halfsel * 8;
#pragma unroll
  for (int t = 0; t < NT; ++t) {
    const int n = nbase + t * 16 + l15;
    const float bv = bias[n];
#pragma unroll
    for (int r = 0; r < 8; ++r) {
      const long rr = row0 + rowadd + r;
      if (rr < M) C[rr * N + n] = acc[t][r] + bv;
    }
  }
}

// ---------------------------------------------------------------------------
// Column statistics: sum and sum-of-squares per column (atomic partials).
// N must divide 256. Launch with grid-stride over rows.
// ---------------------------------------------------------------------------
template <int N>
__global__ void colstats(const float* __restrict__ A, long M,
                         float* __restrict__ sum, float* __restrict__ sumsq) {
  constexpr int RS = 256 / N;
  __shared__ float sh[512];
  const int tid = threadIdx.x;
  const int col = tid & (N - 1);
  const int rsub = tid / N;
  float s = 0.0f, s2 = 0.0f;
  for (long r = (long)blockIdx.x * RS + rsub; r < M; r += (long)gridDim.x * RS) {
    float v = A[r * N + col];
    s += v;
    s2 += v * v;
  }
  sh[tid] = s;
  sh[256 + tid] = s2;
  __syncthreads();
  if (rsub == 0) {
#pragma unroll
    for (int j = 1; j < RS; ++j) {
      s += sh[j * N + col];
      s2 += sh[256 + j * N + col];
    }
    atomicAdd(&sum[col], s);
    atomicAdd(&sumsq[col], s2);
  }
}

// BatchNorm (training stats, biased var) + ReLU, in place.
template <int N>
__global__ void bn_relu(float* __restrict__ A, long M,
                        const float* __restrict__ sum,
                        const float* __restrict__ sumsq,
                        const float* __restrict__ g,
                        const float* __restrict__ b) {
  const long i = (long)blockIdx.x * 256 + threadIdx.x;
  if (i >= M * N) return;
  const int col = (int)(i & (N - 1));
  const float invm = 1.0f / (float)M;
  const float mu = sum[col] * invm;
  const float var = sumsq[col] * invm - mu * mu;
  const float v = (A[i] - mu) * rsqrtf(var + 1e-5f) * g[col] + b[col];
  A[i] = fmaxf(v, 0.0f);
}

// ---------------------------------------------------------------------------
// GENConv softmax aggregation (msg > 0 so uint atomicMax == float max)
// ---------------------------------------------------------------------------
template <int D>
__global__ void msg_max(const float* __restrict__ h, const float* __restrict__ ea,
                        const int* __restrict__ src, const int* __restrict__ dst,
                        float* __restrict__ msg, unsigned* __restrict__ nmax, int E) {
  constexpr int LG = (D == 64) ? 6 : 7;
  const long i = (long)blockIdx.x * 256 + threadIdx.x;
  if (i >= (long)E * D) return;
  const int e = (int)(i >> LG);
  const int d = (int)(i & (D - 1));
  float m = h[(long)src[e] * D + d] + ea[i];
  m = fmaxf(m, 0.0f) + 1e-7f;
  msg[i] = m;
  atomicMax(&nmax[(long)dst[e] * D + d], __float_as_uint(m));
}

template <int D>
__global__ void ex_sum(const float* __restrict__ msg, const int* __restrict__ dst,
                       const unsigned* __restrict__ nmax,
                       float* __restrict__ sex, float* __restrict__ smex, int E) {
  constexpr int LG = (D == 64) ? 6 : 7;
  const long i = (long)blockIdx.x * 256 + threadIdx.x;
  if (i >= (long)E * D) return;
  const int e = (int)(i >> LG);
  const int d = (int)(i & (D - 1));
  const long nd = (long)dst[e] * D + d;
  const float m = msg[i];
  const float ex = expf(m - __uint_as_float(nmax[nd]));
  atomicAdd(&sex[nd], ex);
  atomicAdd(&smex[nd], m * ex);
}

__global__ void aggr_add(const float* __restrict__ sex, const float* __restrict__ smex,
                         const float* __restrict__ h, float* __restrict__ out, long total) {
  const long i = (long)blockIdx.x * 256 + threadIdx.x;
  if (i >= total) return;
  const float se = sex[i];
  const float a = (se > 0.0f) ? (smex[i] / se) : 0.0f;
  out[i] = a + h[i];
}

// ---------------------------------------------------------------------------
// Global mean pool
// ---------------------------------------------------------------------------
__global__ void pool_sum(const float* __restrict__ h, const int* __restrict__ bid,
                         float* __restrict__ gsum, int Nn) {
  const long i = (long)blockIdx.x * 256 + threadIdx.x;
  if (i >= (long)Nn * 64) return;
  const int n = (int)(i >> 6);
  const int d = (int)(i & 63);
  atomicAdd(&gsum[bid[n] * 64 + d], h[i]);
}

__global__ void pool_cnt(const int* __restrict__ bid, float* __restrict__ cnt, int Nn) {
  const int n = blockIdx.x * 256 + threadIdx.x;
  if (n < Nn) atomicAdd(&cnt[bid[n]], 1.0f);
}

// ---------------------------------------------------------------------------
// Output MLP: [64,64] -> 64 (BN,ReLU) -> 32 (BN,ReLU) -> 16 (BN,ReLU) -> 1
// Tiny: single block, 256 threads, LDS-resident.
// ---------------------------------------------------------------------------
__global__ void out_mlp_kernel(
    const float* __restrict__ gsum, const float* __restrict__ cnt,
    const float* w0, const float* b0, const float* g0, const float* bb0,
    const float* w1, const float* b1, const float* g1, const float* bb1,
    const float* w2, const float* b2, const float* g2, const float* bb2,
    const float* w3, const float* b3, float* __restrict__ out) {
  __shared__ float A[64 * 64];
  __shared__ float B[64 * 64];
  __shared__ float mu[64], iv[64];
  const int tid = threadIdx.x;

  for (int i = tid; i < 64 * 64; i += 256)
    A[i] = gsum[i] / fmaxf(cnt[i >> 6], 1.0f);
  __syncthreads();

  // L0: 64 -> 64
  for (int i = tid; i < 64 * 64; i += 256) {
    const int r = i >> 6, c = i & 63;
    float s = b0[c];
    for (int k = 0; k < 64; ++k) s += A[r * 64 + k] * w0[k * 64 + c];
    B[i] = s;
  }
  __syncthreads();
  if (tid < 64) {
    float s = 0, s2 = 0;
    for (int r = 0; r < 64; ++r) { float v = B[r * 64 + tid]; s += v; s2 += v * v; }
    const float m = s / 64.0f;
    mu[tid] = m;
    iv[tid] = rsqrtf(s2 / 64.0f - m * m + 1e-5f);
  }
  __syncthreads();
  for (int i = tid; i < 64 * 64; i += 256) {
    const int c = i & 63;
    B[i] = fmaxf((B[i] - mu[c]) * iv[c] * g0[c] + bb0[c], 0.0f);
  }
  __syncthreads();

  // L1: 64 -> 32
  for (int i = tid; i < 64 * 32; i += 256) {
    const int r = i >> 5, c = i & 31;
    float s = b1[c];
    for (int k = 0; k < 64; ++k) s += B[r * 64 + k] * w1[k * 32 + c];
    A[i] = s;
  }
  __syncthreads();
  if (tid < 32) {
    float s = 0, s2 = 0;
    for (int r = 0; r < 64; ++r) { float v = A[r * 32 + tid]; s += v; s2 += v * v; }
    const float m = s / 64.0f;
    mu[tid] = m;
    iv[tid] = rsqrtf(s2 / 64.0f - m * m + 1e-5f);
  }
  __syncthreads();
  for (int i = tid; i < 64 * 32; i += 256) {
    const int c = i & 31;
    A[i] = fmaxf((A[i] - mu[c]) * iv[c] * g1[c] + bb1[c], 0.0f);
  }
  __syncthreads();

  // L2: 32 -> 16
  for (int i = tid; i < 64 * 16; i += 256) {
    const int r = i >> 4, c = i & 15;
    float s = b2[c];
    for (int k = 0; k < 32; ++k) s += A[r * 32 + k] * w2[k * 16 + c];
    B[i] = s;
  }
  __syncthreads();
  if (tid < 16) {
    float s = 0, s2 = 0;
    for (int r = 0; r < 64; ++r) { float v = B[r * 16 + tid]; s += v; s2 += v * v; }
    const float m = s / 64.0f;
    mu[tid] = m;
    iv[tid] = rsqrtf(s2 / 64.0f - m * m + 1e-5f);
  }
  __syncthreads();
  for (int i = tid; i < 64 * 16; i += 256) {
    const int c = i & 15;
    B[i] = fmaxf((B[i] - mu[c]) * iv[c] * g2[c] + bb2[c], 0.0f);
  }
  __syncthreads();

  // L3: 16 -> 1 (plain)
  if (tid < 64) {
    float s = b3[0];
    for (int k = 0; k < 16; ++k) s += B[tid * 16 + k] * w3[k];
    out[tid] = s;
  }
}

// ---------------------------------------------------------------------------
// Host-side helpers
// ---------------------------------------------------------------------------
struct Lyr { const float *w, *b, *g, *bb; };

template <int N>
static void stats_bn(float* data, long M, const float* g, const float* bb,
                     float* sum, float* sumsq, hipStream_t stream) {
  hipMemsetAsync(sum, 0, 512 * sizeof(float), stream);  // sum+sumsq contiguous
  colstats<N><<<512, 256, 0, stream>>>(data, M, sum, sumsq);
  const long tot = M * N;
  const int blks = (int)((tot + 255) / 256);
  bn_relu<N><<<blks, 256, 0, stream>>>(data, M, sum, sumsq, g, bb);
}

template <int D>
static void conv_aggregate(const float* hin, const float* eaD,
                           const int* src, const int* dst,
                           float* msgbuf, unsigned* nm, float* sex, float* smex,
                           float* aggout, int Nn, int E, hipStream_t stream) {
  hipMemsetAsync(nm, 0, (size_t)Nn * D * sizeof(float), stream);
  hipMemsetAsync(sex, 0, (size_t)Nn * D * sizeof(float), stream);
  hipMemsetAsync(smex, 0, (size_t)Nn * D * sizeof(float), stream);
  const long tot = (long)E * D;
  const int blks = (int)((tot + 255) / 256);
  msg_max<D><<<blks, 256, 0, stream>>>(hin, eaD, src, dst, msgbuf, nm, E);
  ex_sum<D><<<blks, 256, 0, stream>>>(msgbuf, dst, nm, sex, smex, E);
  const long ntot = (long)Nn * D;
  const int nblks = (int)((ntot + 255) / 256);
  aggr_add<<<nblks, 256, 0, stream>>>(sex, smex, hin, aggout, ntot);
}

extern "C" void kernel_launch(void* const* d_in, const int* in_sizes, int n_in,
                              void* d_out, int out_size, void* d_ws, size_t ws_size,
                              hipStream_t stream) {
  (void)n_in; (void)out_size; (void)ws_size;
  const int E = NEDGES, Nn = NNODES;

  const float* x  = (const float*)d_in[0];
  const float* ea_in = (const float*)d_in[1];
  const int* eidx = (const int*)d_in[2];
  const int* src = eidx;
  const int* dst = eidx + E;
  const int* bid = (const int*)d_in[3];

  // Detect param flattening order:
  //   JAX sorted-key pytree: idx4 = conv_h1_mlp L0 bn.b (size 128)
  //   dict insertion order : idx4 = edge_mlp_in L0 lin.w (size 16384)
  const bool sorted_order = (in_sizes[4] == 128);
  auto layer = [&](int base, int l) -> Lyr {
    const int i = base + l * 4;
    Lyr p;
    if (sorted_order) {  // [bn.b, bn.g, lin.b, lin.w]
      p.bb = (const float*)d_in[i];
      p.g  = (const float*)d_in[i + 1];
      p.b  = (const float*)d_in[i + 2];
      p.w  = (const float*)d_in[i + 3];
    } else {             // [lin.w, lin.b, bn.g, bn.b]
      p.w  = (const float*)d_in[i];
      p.b  = (const float*)d_in[i + 1];
      p.g  = (const float*)d_in[i + 2];
      p.bb = (const float*)d_in[i + 3];
    }
    return p;
  };
  const int b_h1 = sorted_order ? 4 : 28;
  const int b_h2 = sorted_order ? 12 : 36;
  const int b_ci = sorted_order ? 20 : 12;
  const int b_eh = sorted_order ? 28 : 20;
  const int b_ei = sorted_order ? 36 : 4;
  const int b_ou = 44;

  Lyr ei0 = layer(b_ei, 0), ei1 = layer(b_ei, 1);
  Lyr ci0 = layer(b_ci, 0), ci1 = layer(b_ci, 1);
  Lyr eh0 = layer(b_eh, 0), eh1 = layer(b_eh, 1);
  Lyr h10 = layer(b_h1, 0), h11 = layer(b_h1, 1);
  Lyr h20 = layer(b_h2, 0), h21 = layer(b_h2, 1);
  Lyr o0 = layer(b_ou, 0), o1 = layer(b_ou, 1), o2 = layer(b_ou, 2), o3 = layer(b_ou, 3);

  // Workspace layout
  size_t off = 0;
  auto alloc = [&](size_t bytes) -> char* {
    char* p = (char*)d_ws + off;
    off += (bytes + 255) & ~(size_t)255;
    return p;
  };
  float* EB0   = (float*)alloc((size_t)E * 128 * 4);   // edge buf (ea / ea2)
  float* EB1   = (float*)alloc((size_t)E * 128 * 4);   // edge buf (tmp / msg)
  float* nm    = (float*)alloc((size_t)Nn * 128 * 4);  // seg max bits
  float* sex   = (float*)alloc((size_t)Nn * 128 * 4);  // seg sum exp
  float* smex  = (float*)alloc((size_t)Nn * 128 * 4);  // seg sum msg*exp
  float* aggin = (float*)alloc((size_t)Nn * 128 * 4);  // aggr + h
  float* hid   = (float*)alloc((size_t)Nn * 256 * 4);  // conv hidden
  float* hA    = (float*)alloc((size_t)Nn * 64 * 4);   // node feats
  float* hB    = (float*)alloc((size_t)Nn * 64 * 4);
  float* stats = (float*)alloc(512 * 4);               // sum[256] ++ sumsq[256]
  float* gsum  = (float*)alloc(64 * 64 * 4);
  float* gcnt  = (float*)alloc(64 * 4);
  float* sum = stats, *sumsq = stats + 256;

  const int geb = E / 128;                  // 3125
  const int gnb = (Nn + 127) / 128;         // 391

  // ---- ea = edge_mlp_in(edge_attr) : 128->128 (BN,ReLU) ->128 ----
  gemm_bias<128, 128><<<dim3(geb, 2), 256, 0, stream>>>(ea_in, ei0.w, ei0.b, EB1, E);
  stats_bn<128>(EB1, E, ei0.g, ei0.bb, sum, sumsq, stream);
  gemm_bias<128, 128><<<dim3(geb, 2), 256, 0, stream>>>(EB1, ei1.w, ei1.b, EB0, E);

  // ---- conv_in: aggregate(x, ea) -> MLP 128->256 (BN,ReLU) ->64 ----
  conv_aggregate<128>(x, EB0, src, dst, EB1, (unsigned*)nm, sex, smex, aggin, Nn, E, stream);
  gemm_bias<128, 256><<<dim3(gnb, 4), 256, 0, stream>>>(aggin, ci0.w, ci0.b, hid, Nn);
  stats_bn<256>(hid, Nn, ci0.g, ci0.bb, sum, sumsq, stream);
  gemm_bias<256, 64><<<dim3(gnb, 1), 256, 0, stream>>>(hid, ci1.w, ci1.b, hA, Nn);

  // ---- ea2 = edge_mlp_hid(edge_attr) : 128->64 (BN,ReLU) ->64 ----
  gemm_bias<128, 64><<<dim3(geb, 1), 256, 0, stream>>>(ea_in, eh0.w, eh0.b, EB1, E);
  stats_bn<64>(EB1, E, eh0.g, eh0.bb, sum, sumsq, stream);
  gemm_bias<64, 64><<<dim3(geb, 1), 256, 0, stream>>>(EB1, eh1.w, eh1.b, EB0, E);

  // ---- conv_h1 ----
  conv_aggregate<64>(hA, EB0, src, dst, EB1, (unsigned*)nm, sex, smex, aggin, Nn, E, stream);
  gemm_bias<64, 128><<<dim3(gnb, 2), 256, 0, stream>>>(aggin, h10.w, h10.b, hid, Nn);
  stats_bn<128>(hid, Nn, h10.g, h10.bb, sum, sumsq, stream);
  gemm_bias<128, 64><<<dim3(gnb, 1), 256, 0, stream>>>(hid, h11.w, h11.b, hB, Nn);

  // ---- conv_h2 ----
  conv_aggregate<64>(hB, EB0, src, dst, EB1, (unsigned*)nm, sex, smex, aggin, Nn, E, stream);
  gemm_bias<64, 128><<<dim3(gnb, 2), 256, 0, stream>>>(aggin, h20.w, h20.b, hid, Nn);
  stats_bn<128>(hid, Nn, h20.g, h20.bb, sum, sumsq, stream);
  gemm_bias<128, 64><<<dim3(gnb, 1), 256, 0, stream>>>(hid, h21.w, h21.b, hA, Nn);

  // ---- global mean pool + output MLP ----
  hipMemsetAsync(gsum, 0, 64 * 64 * sizeof(float), stream);
  hipMemsetAsync(gcnt, 0, 64 * sizeof(float), stream);
  pool_sum<<<(Nn * 64 + 255) / 256, 256, 0, stream>>>(hA, bid, gsum, Nn);
  pool_cnt<<<(Nn + 255) / 256, 256, 0, stream>>>(bid, gcnt, Nn);
  out_mlp_kernel<<<1, 256, 0, stream>>>(gsum, gcnt,
      o0.w, o0.b, o0.g, o0.bb, o1.w, o1.b, o1.g, o1.bb,
      o2.w, o2.b, o2.g, o2.bb, o3.w, o3.b, (float*)d_out);
}